// GBiNet_420906795177
// MI455X (gfx1250) — compile-verified
//
#include <hip/hip_runtime.h>

typedef float v2f __attribute__((ext_vector_type(2)));
typedef float v8f __attribute__((ext_vector_type(8)));

#define KB 2
#define KV 5
#define KC 32
#define KH 256
#define KW 320
#define KD 4
#define KG 8
#define KHW (KH * KW)

// Compiler-level fence for per-wave LDS staging: CDNA5 keeps same-wave LDS ops
// in order (DScnt, in-order), so no hardware workgroup barrier is needed —
// only prevent the compiler from reordering the DS ops. The workgroup fence
// lowers to (at most) s_wait_dscnt, which the dependent loads need anyway.
__device__ __forceinline__ void wave_lds_fence() {
  __builtin_amdgcn_fence(__ATOMIC_ACQ_REL, "workgroup");
  __builtin_amdgcn_wave_barrier();
}

__device__ __forceinline__ void inv3x3(const float* m, float* o) {
  float a = m[0], b = m[1], c = m[2];
  float d = m[3], e = m[4], f = m[5];
  float g = m[6], h = m[7], i = m[8];
  float c00 = (e * i - f * h);
  float c01 = -(d * i - f * g);
  float c02 = (d * h - e * g);
  float det = a * c00 + b * c01 + c * c02;
  float r = 1.0f / det;
  o[0] = (e * i - f * h) * r;
  o[1] = -(b * i - c * h) * r;
  o[2] = (b * f - c * e) * r;
  o[3] = -(d * i - f * g) * r;
  o[4] = (a * i - c * g) * r;
  o[5] = -(a * f - c * d) * r;
  o[6] = (d * h - e * g) * r;
  o[7] = -(a * h - b * g) * r;
  o[8] = (a * e - b * d) * r;
}

#define WMMAF32(A, Bv, C) \
  __builtin_amdgcn_wmma_f32_16x16x4_f32(false, (A), false, (Bv), (short)0, (C), false, false)

__global__ __launch_bounds__(256) void gbinet_kernel(
    const float* __restrict__ feat, const float* __restrict__ depths,
    const float* __restrict__ intr, const float* __restrict__ extr,
    const float* __restrict__ w0, const float* __restrict__ b0,
    const float* __restrict__ w1, const float* __restrict__ b1,
    const float* __restrict__ w2, const float* __restrict__ b2,
    float* __restrict__ out) {
  __shared__ __attribute__((aligned(16))) float sW0[128];
  __shared__ __attribute__((aligned(16))) float sW1[128];
  __shared__ __attribute__((aligned(16))) float sW2[8];
  __shared__ __attribute__((aligned(16))) float sB0[16];
  __shared__ __attribute__((aligned(16))) float sB1[8];
  __shared__ float sB2[1];
  // per-wave staging buffers (no cross-wave traffic -> no block barriers)
  __shared__ __attribute__((aligned(16))) float sSim[8][128 * 8];  // [col][g]
  __shared__ __attribute__((aligned(16))) float sX1[8][16 * 16];   // [locol][row]
  __shared__ __attribute__((aligned(16))) float sX2[8][16 * 16];   // [locol][row] + pad half
  __shared__ __attribute__((aligned(16))) float sO[8][128];        // [col]

  const int tid = threadIdx.x;
  const int lane = tid & 31;
  const int wv = tid >> 5;

  if (tid < 128) { sW0[tid] = w0[tid]; sW1[tid] = w1[tid]; }
  if (tid < 16) sB0[tid] = b0[tid];
  if (tid < 8) { sW2[tid] = w2[tid]; sB1[tid] = b1[tid]; }
  if (tid == 0) sB2[0] = b2[0];
  __syncthreads();  // only true cross-wave dependency in the kernel

  const int p = blockIdx.x * 256 + tid;
  const int b = p / KHW;
  const int hw = p % KHW;
  const int h = hw / KW;
  const int w = hw % KW;
  const int bu = __builtin_amdgcn_readfirstlane(b);

  // ---- reference camera: K_ref_inv, R_ref_inv, t0 ----
  float Kin[9], Rin[9], t0[3];
  {
    float Km[9], Rm[9];
    const float* K0 = intr + (size_t)(bu * KV) * 9;
    const float* E0 = extr + (size_t)(bu * KV) * 12;
#pragma unroll
    for (int i = 0; i < 9; i++) Km[i] = K0[i];
#pragma unroll
    for (int r = 0; r < 3; r++) {
      Rm[r * 3 + 0] = E0[r * 4 + 0];
      Rm[r * 3 + 1] = E0[r * 4 + 1];
      Rm[r * 3 + 2] = E0[r * 4 + 2];
    }
    inv3x3(Km, Kin);
    inv3x3(Rm, Rin);
    t0[0] = E0[3]; t0[1] = E0[7]; t0[2] = E0[11];
  }

  // ---- world points for the 4 depth hypotheses ----
  const float gx = (float)w + 0.5f;
  const float gy = (float)h + 0.5f;
  const float ux = Kin[0] * gx + Kin[1] * gy + Kin[2];
  const float uy = Kin[3] * gx + Kin[4] * gy + Kin[5];
  const float uz = Kin[6] * gx + Kin[7] * gy + Kin[8];
  float wxp[KD], wyp[KD], wzp[KD];
#pragma unroll
  for (int d = 0; d < KD; ++d) {
    float dep = depths[(((size_t)b * KD + d) * KH + h) * KW + w];
    float cx = ux * dep - t0[0];
    float cy = uy * dep - t0[1];
    float cz = uz * dep - t0[2];
    wxp[d] = Rin[0] * cx + Rin[1] * cy + Rin[2] * cz;
    wyp[d] = Rin[3] * cx + Rin[4] * cy + Rin[5] * cz;
    wzp[d] = Rin[6] * cx + Rin[7] * cy + Rin[8] * cz;
  }

  // ---- reference features (view 0), 32 channels ----
  float ref[KC];
  {
    const float* fr = feat + ((size_t)b * KV) * KC * KHW + hw;
#pragma unroll
    for (int c = 0; c < KC; c++) ref[c] = fr[(size_t)c * KHW];
  }

  // ---- WMMA A-fragments (16x4 f32 layout: M=lane%16; VGPR0/1 = K k0,k0+1
  //      with k0 = 0 for lanes 0-15, 2 for lanes 16-31) ----
  const int m16 = lane & 15;
  const int kk0 = (lane < 16) ? 0 : 2;
  v2f aW0[2], aW1[4], aW2[2];
#pragma unroll
  for (int kk = 0; kk < 2; ++kk) {
    aW0[kk].x = sW0[m16 * 8 + kk * 4 + kk0];
    aW0[kk].y = sW0[m16 * 8 + kk * 4 + kk0 + 1];
  }
#pragma unroll
  for (int kk = 0; kk < 4; ++kk) {
    if (m16 < 8) {
      aW1[kk].x = sW1[m16 * 16 + kk * 4 + kk0];
      aW1[kk].y = sW1[m16 * 16 + kk * 4 + kk0 + 1];
    } else {
      aW1[kk].x = 0.0f; aW1[kk].y = 0.0f;
    }
  }
#pragma unroll
  for (int kk = 0; kk < 2; ++kk) {
    if ((m16 & 7) == 0) {  // rows 0 and 8 replicate W2 so both lane halves get D row in VGPR0
      aW2[kk].x = sW2[kk * 4 + kk0];
      aW2[kk].y = sW2[kk * 4 + kk0 + 1];
    } else {
      aW2[kk].x = 0.0f; aW2[kk].y = 0.0f;
    }
  }
  float bia0[8], bia1[8];
#pragma unroll
  for (int r = 0; r < 8; r++) {
    bia0[r] = sB0[((lane < 16) ? 0 : 8) + r];
    bia1[r] = sB1[r];
  }
  const float bia2 = sB2[0];

  float wsum = 0.0f;
  float simsum[32];
#pragma unroll
  for (int i = 0; i < 32; i++) simsum[i] = 0.0f;

  float* Sw = sSim[wv];
  float* X1w = sX1[wv];
  float* X2w = sX2[wv];
  float* Ow = sO[wv];

  // layer-1 store: lanes >= 16 hold the zero-padded rows 8..15 -> park them in
  // the scratch half of X2w so the store is EXEC-uniform (no saveexec/branch).
  const int x2base = m16 * 8 + ((lane < 16) ? 0 : 128);

#pragma unroll 1
  for (int s = 1; s < KV; ++s) {
    // proj = (Ks*Rs) @ world + Ks*ts
    const float* Ks = intr + (size_t)(bu * KV + s) * 9;
    const float* Es = extr + (size_t)(bu * KV + s) * 12;
    float M[9], q[3];
#pragma unroll
    for (int r = 0; r < 3; r++) {
#pragma unroll
      for (int c2 = 0; c2 < 3; c2++)
        M[r * 3 + c2] = Ks[r * 3 + 0] * Es[0 * 4 + c2] + Ks[r * 3 + 1] * Es[1 * 4 + c2] +
                        Ks[r * 3 + 2] * Es[2 * 4 + c2];
      q[r] = Ks[r * 3 + 0] * Es[3] + Ks[r * 3 + 1] * Es[7] + Ks[r * 3 + 2] * Es[11];
    }
    const float* fs = feat + ((size_t)(b * KV + s)) * KC * KHW;

#pragma unroll 1
    for (int d = 0; d < KD; ++d) {
      float X = M[0] * wxp[d] + M[1] * wyp[d] + M[2] * wzp[d] + q[0];
      float Y = M[3] * wxp[d] + M[4] * wyp[d] + M[5] * wzp[d] + q[1];
      float Z = M[6] * wxp[d] + M[7] * wyp[d] + M[8] * wzp[d] + q[2];
      float z = fmaxf(Z, 1e-6f);
      float inz = 1.0f / z;
      float px = X * inz, py = Y * inz;
      float x0f = floorf(px), y0f = floorf(py);
      float fx = px - x0f, fy = py - y0f;
      float x1f = x0f + 1.0f, y1f = y0f + 1.0f;
      float vx0 = (x0f >= 0.0f && x0f <= (float)(KW - 1)) ? 1.0f : 0.0f;
      float vx1 = (x1f >= 0.0f && x1f <= (float)(KW - 1)) ? 1.0f : 0.0f;
      float vy0 = (y0f >= 0.0f && y0f <= (float)(KH - 1)) ? 1.0f : 0.0f;
      float vy1 = (y1f >= 0.0f && y1f <= (float)(KH - 1)) ? 1.0f : 0.0f;
      float w00 = (1.0f - fx) * (1.0f - fy) * vx0 * vy0;
      float w01 = fx * (1.0f - fy) * vx1 * vy0;
      float w10 = (1.0f - fx) * fy * vx0 * vy1;
      float w11 = fx * fy * vx1 * vy1;
      int xi0 = (int)fminf(fmaxf(x0f, 0.0f), (float)(KW - 1));
      int xi1 = (int)fminf(fmaxf(x1f, 0.0f), (float)(KW - 1));
      int yi0 = (int)fminf(fmaxf(y0f, 0.0f), (float)(KH - 1));
      int yi1 = (int)fminf(fmaxf(y1f, 0.0f), (float)(KH - 1));
      int i00 = yi0 * KW + xi0, i01 = yi0 * KW + xi1;
      int i10 = yi1 * KW + xi0, i11 = yi1 * KW + xi1;

      float simg[KG];
#pragma unroll
      for (int g = 0; g < KG; ++g) {
        float acc = 0.0f;
#pragma unroll
        for (int j = 0; j < 4; ++j) {
          int c = g * 4 + j;
          const float* fc = fs + (size_t)c * KHW;
          float sv = w00 * fc[i00] + w01 * fc[i01] + w10 * fc[i10] + w11 * fc[i11];
          acc += sv * ref[c];
        }
        simg[g] = 0.25f * acc;
      }
      float4* dstS = (float4*)&Sw[(lane * 4 + d) * 8];
      dstS[0] = make_float4(simg[0], simg[1], simg[2], simg[3]);
      dstS[1] = make_float4(simg[4], simg[5], simg[6], simg[7]);
    }
    wave_lds_fence();

    // ---- per-wave MLP over 128 columns, 8 tiles of N=16, via f32 WMMA ----
#pragma unroll 1
    for (int t = 0; t < 8; ++t) {
      const int cl = m16;
      const int cg = t * 16 + cl;
      // layer 0: 16x8 @ 8xN
      v8f acc0 = {0.f, 0.f, 0.f, 0.f, 0.f, 0.f, 0.f, 0.f};
#pragma unroll
      for (int kk = 0; kk < 2; ++kk) {
        v2f bb = *(const v2f*)&Sw[cg * 8 + kk * 4 + kk0];
        acc0 = WMMAF32(aW0[kk], bb, acc0);
      }
      {
        float x1v[8];
#pragma unroll
        for (int r = 0; r < 8; r++) x1v[r] = fmaxf(acc0[r] + bia0[r], 0.0f);
        float4* d4 = (float4*)&X1w[cl * 16 + ((lane < 16) ? 0 : 8)];
        d4[0] = make_float4(x1v[0], x1v[1], x1v[2], x1v[3]);
        d4[1] = make_float4(x1v[4], x1v[5], x1v[6], x1v[7]);
      }
      wave_lds_fence();
      // layer 1: 8x16 @ 16xN (rows 8-15 of A zero-padded)
      v8f acc1 = {0.f, 0.f, 0.f, 0.f, 0.f, 0.f, 0.f, 0.f};
#pragma unroll
      for (int kk = 0; kk < 4; ++kk) {
        v2f bb = *(const v2f*)&X1w[cl * 16 + kk * 4 + kk0];
        acc1 = WMMAF32(aW1[kk], bb, acc1);
      }
      {
        float x2v[8];
#pragma unroll
        for (int r = 0; r < 8; r++) x2v[r] = fmaxf(acc1[r] + bia1[r], 0.0f);
        float4* d4 = (float4*)&X2w[x2base];  // lanes>=16 write scratch half
        d4[0] = make_float4(x2v[0], x2v[1], x2v[2], x2v[3]);
        d4[1] = make_float4(x2v[4], x2v[5], x2v[6], x2v[7]);
      }
      wave_lds_fence();
      // layer 2: 1x8 @ 8xN, W2 in A rows 0 and 8 -> every lane's VGPR0 = its column's output
      v8f acc2 = {0.f, 0.f, 0.f, 0.f, 0.f, 0.f, 0.f, 0.f};
#pragma unroll
      for (int kk = 0; kk < 2; ++kk) {
        v2f bb = *(const v2f*)&X2w[cl * 8 + kk * 4 + kk0];
        acc2 = WMMAF32(aW2[kk], bb, acc2);
      }
      float ov = acc2[0] + bia2;
      ov = 1.0f / (1.0f + __expf(-ov));
      Ow[cg] = ov;  // lanes l and l+16 write identical values -> benign, EXEC-uniform
      wave_lds_fence();
    }

    // ---- view weight = max over D, accumulate ----
    float o0 = Ow[lane * 4 + 0];
    float o1 = Ow[lane * 4 + 1];
    float o2 = Ow[lane * 4 + 2];
    float o3 = Ow[lane * 4 + 3];
    float vw = fmaxf(fmaxf(o0, o1), fmaxf(o2, o3));
    wsum += vw;
#pragma unroll
    for (int d = 0; d < KD; ++d) {
      const float4 s0 = *(const float4*)&Sw[(lane * 4 + d) * 8];
      const float4 s1 = *(const float4*)&Sw[(lane * 4 + d) * 8 + 4];
      simsum[d * 8 + 0] += vw * s0.x;
      simsum[d * 8 + 1] += vw * s0.y;
      simsum[d * 8 + 2] += vw * s0.z;
      simsum[d * 8 + 3] += vw * s0.w;
      simsum[d * 8 + 4] += vw * s1.x;
      simsum[d * 8 + 5] += vw * s1.y;
      simsum[d * 8 + 6] += vw * s1.z;
      simsum[d * 8 + 7] += vw * s1.w;
    }
    wave_lds_fence();  // before next view overwrites Sw
  }

  // ---- write out [B,G,D,H,W] ----
  const float invw = 1.0f / wsum;
#pragma unroll
  for (int g = 0; g < KG; ++g) {
#pragma unroll
    for (int d = 0; d < KD; ++d) {
      out[(((size_t)b * KG + g) * KD + d) * KHW + hw] = simsum[d * 8 + g] * invw;
    }
  }
}

extern "C" void kernel_launch(void* const* d_in, const int* in_sizes, int n_in,
                              void* d_out, int out_size, void* d_ws, size_t ws_size,
                              hipStream_t stream) {
  (void)in_sizes; (void)n_in; (void)out_size; (void)d_ws; (void)ws_size;
  const float* feat = (const float*)d_in[0];
  const float* depths = (const float*)d_in[1];
  // d_in[2] = feature_map_indices_grid (recomputed analytically in-kernel)
  const float* intr = (const float*)d_in[3];
  const float* extr = (const float*)d_in[4];
  const float* w0 = (const float*)d_in[5];
  const float* b0 = (const float*)d_in[6];
  const float* w1 = (const float*)d_in[7];
  const float* b1 = (const float*)d_in[8];
  const float* w2 = (const float*)d_in[9];
  const float* b2 = (const float*)d_in[10];
  float* out = (float*)d_out;

  dim3 grid((KB * KHW) / 256);  // 640 blocks, exact cover -> EXEC all-1s for WMMA
  gbinet_kernel<<<grid, 256, 0, stream>>>(feat, depths, intr, extr, w0, b0, w1, b1, w2, b2, out);
}